// WER_63170378989842
// MI455X (gfx1250) — compile-verified
//
#include <hip/hip_runtime.h>
#include <stdint.h>

// Problem shape (fixed by reference): B=16, S=T=512, V=32000.
static constexpr int kVocab = 32000;
static constexpr int kSeq   = 512;
// Async staging: 256 threads * float4 = 1024 floats = 4096 B per stage.
static constexpr int kStageFloats = 1024;
static constexpr int kStageBytes  = 4096;
static constexpr int kFullStages  = 31;        // 31*1024 = 31744
static constexpr int kRemBase     = 31744;     // remaining 256 floats -> 1/thread
static constexpr int kNBuf        = 4;         // 4-deep async pipeline
static constexpr int kSteady      = 24;        // multiple of kNBuf; drain = 7

// Branchless argmax update: strict '>' keeps first occurrence; tie -> lower
// index (jnp.argmax rule). Lowers to v_cmp + v_cndmask, no control flow.
__device__ __forceinline__ void upd(float v, int i, float& bv, int& bi) {
    const bool better = (v > bv) | ((v == bv) & (i < bi));
    bv = better ? v : bv;
    bi = better ? i : bi;
}

// ---------------------------------------------------------------------------
// Kernel 1: per-(b,s)-row argmax over 32000 logits.
// HBM-bound streaming via CDNA5 async global->LDS (ASYNCcnt), 4-deep
// pipeline to build up in-flight bytes (latency*BW product). Each lane copies
// & consumes its own 16B LDS slot => no workgroup barriers in the hot loop,
// only the wave-local s_wait_asynccnt. Steady-state loop is branch-free with
// constant wait immediates; tail stages drained with literal counts.
// ---------------------------------------------------------------------------
__global__ __launch_bounds__(256) void wer_argmax_kernel(
        const float* __restrict__ logits, int* __restrict__ words) {
    const int row = blockIdx.x;           // 0 .. B*S-1
    const int tid = threadIdx.x;          // 0 .. 255
    const uint64_t gbase = (uint64_t)(logits + (size_t)row * kVocab);

    __shared__ float4 stage[kNBuf][256];
    __shared__ float  rv[256];
    __shared__ int    ri[256];

    // Low 32 bits of the generic pointer == workgroup-relative LDS byte offset.
    uint32_t ldsoff[kNBuf];
#pragma unroll
    for (int p = 0; p < kNBuf; ++p)
        ldsoff[p] = (uint32_t)(uintptr_t)&stage[p][tid];

    float bv = -3.402823466e38f;
    int   bi = 0x7fffffff;

    auto consume = [&](int s) {
        float4 v = stage[s & (kNBuf - 1)][tid];                // ds_load_b128
        int idx0 = s * kStageFloats + tid * 4;
        upd(v.x, idx0 + 0, bv, bi);
        upd(v.y, idx0 + 1, bv, bi);
        upd(v.z, idx0 + 2, bv, bi);
        upd(v.w, idx0 + 3, bv, bi);
    };
    auto issue = [&](int s, int buf) {
        uint32_t goff = (uint32_t)(s * kStageBytes) + (uint32_t)(tid * 16u);
        // s_wait_dscnt 0 first: our ds_load of this buffer must have drained
        // before the async engine may overwrite it (WAR on LDS).
        asm volatile("s_wait_dscnt 0x0\n\t"
                     "global_load_async_to_lds_b128 %0, %1, %2 offset:0 th:TH_LOAD_NT"
                     :: "v"(ldsoff[buf]), "v"(goff), "s"(gbase) : "memory");
    };

    // Prime the pipeline: stages 0..3. NT hint: read-once 1GB stream,
    // keep it out of the 192MB L2's way.
#pragma unroll
    for (int p = 0; p < kNBuf; ++p) {
        uint32_t goff = (uint32_t)(p * kStageBytes) + (uint32_t)(tid * 16u);
        asm volatile("global_load_async_to_lds_b128 %0, %1, %2 offset:0 th:TH_LOAD_NT"
                     :: "v"(ldsoff[p]), "v"(goff), "s"(gbase) : "memory");
    }

    // Tail (elements 31744..31999): one direct NT load per thread.
    float remv = __builtin_nontemporal_load(
        (const float*)gbase + (kRemBase + tid));

    // Steady state: always 4 stages in flight, constant wait immediate.
#pragma unroll 4
    for (int s = 0; s < kSteady; ++s) {
        asm volatile("s_wait_asynccnt 0x3" ::: "memory");  // stage s landed
        consume(s);
        issue(s + kNBuf, s & (kNBuf - 1));
    }
    // Drain stages 24..30 (issue the final stages 28..30 along the way).
    asm volatile("s_wait_asynccnt 0x3" ::: "memory"); consume(24); issue(28, 0);
    asm volatile("s_wait_asynccnt 0x3" ::: "memory"); consume(25); issue(29, 1);
    asm volatile("s_wait_asynccnt 0x3" ::: "memory"); consume(26); issue(30, 2);
    asm volatile("s_wait_asynccnt 0x3" ::: "memory"); consume(27);
    asm volatile("s_wait_asynccnt 0x2" ::: "memory"); consume(28);
    asm volatile("s_wait_asynccnt 0x1" ::: "memory"); consume(29);
    asm volatile("s_wait_asynccnt 0x0" ::: "memory"); consume(30);

    upd(remv, kRemBase + tid, bv, bi);

    // Workgroup (val,idx) reduction, branchless combine, same tie-break rule.
    rv[tid] = bv; ri[tid] = bi;
    __syncthreads();
    for (int off = 128; off > 0; off >>= 1) {
        if (tid < off) {
            float ov = rv[tid + off]; int oi = ri[tid + off];
            float cv = rv[tid];       int ci = ri[tid];
            const bool better = (ov > cv) | ((ov == cv) & (oi < ci));
            rv[tid] = better ? ov : cv;
            ri[tid] = better ? oi : ci;
        }
        __syncthreads();
    }
    if (tid == 0) words[row] = ri[0];
}

// ---------------------------------------------------------------------------
// Kernel 2: Levenshtein DP, anti-diagonal wavefront. One block per batch item.
// dbuf holds 3 rotating diagonals; exact replica of the reference recurrence:
// match -> diag; else min(up,left,diag)+1.
// ---------------------------------------------------------------------------
__global__ __launch_bounds__(512) void wer_editdist_kernel(
        const int* __restrict__ words, const int* __restrict__ target,
        float* __restrict__ dists) {
    const int b   = blockIdx.x;
    const int tid = threadIdx.x;      // 0..511
    __shared__ int ow[kSeq];
    __shared__ int tg[kSeq];
    __shared__ int dbuf[3][kSeq + 2];

    ow[tid] = words[b * kSeq + tid];
    tg[tid] = target[b * kSeq + tid];
    __syncthreads();

    for (int k = 0; k <= 2 * kSeq; ++k) {
        const int cb = k % 3;              // diagonal k
        const int p1 = (k + 2) % 3;        // diagonal k-1 (up / left)
        const int p2 = (k + 1) % 3;        // diagonal k-2 (diag)
        for (int j = tid; j <= kSeq; j += 512) {   // tid 0 also covers j=512
            if (j <= k && (k - j) <= kSeq) {
                const int i = k - j;
                int val;
                if (i == 0)       val = j;
                else if (j == 0)  val = i;
                else {
                    int up   = dbuf[p1][j];
                    int left = dbuf[p1][j - 1];
                    int dg   = dbuf[p2][j - 1];
                    if (ow[i - 1] == tg[j - 1]) val = dg;
                    else {
                        int m = up < left ? up : left;
                        m = m < dg ? m : dg;
                        val = m + 1;
                    }
                }
                dbuf[cb][j] = val;
            }
        }
        __syncthreads();
    }
    if (tid == 0) dists[b] = (float)dbuf[(2 * kSeq) % 3][kSeq];
}

// ---------------------------------------------------------------------------
// Kernel 3: deterministic scalar reduction: mean(dist / T) over batch.
// ---------------------------------------------------------------------------
__global__ void wer_finalize_kernel(const float* __restrict__ dists,
                                    float* __restrict__ out, int B) {
    if (threadIdx.x == 0 && blockIdx.x == 0) {
        float s = 0.0f;
        for (int b = 0; b < B; ++b) s += dists[b];
        out[0] = s / ((float)B * (float)kSeq);
    }
}

extern "C" void kernel_launch(void* const* d_in, const int* in_sizes, int n_in,
                              void* d_out, int out_size, void* d_ws, size_t ws_size,
                              hipStream_t stream) {
    const float* logits = (const float*)d_in[0];  // (B, S, V) f32
    const int*   target = (const int*)d_in[1];    // (B, S) i32
    const int BS = in_sizes[1];                   // B * S rows
    const int B  = BS / kSeq;

    int*   words = (int*)d_ws;                                  // BS ints
    float* dists = (float*)((char*)d_ws + (size_t)BS * sizeof(int)); // B floats

    wer_argmax_kernel  <<<BS, 256, 0, stream>>>(logits, words);
    wer_editdist_kernel<<<B,  kSeq, 0, stream>>>(words, target, dists);
    wer_finalize_kernel<<<1,  64,  0, stream>>>(dists, (float*)d_out, B);
}